// Elastodynamics_28810640622164
// MI455X (gfx1250) — compile-verified
//
#include <hip/hip_runtime.h>

// ---------------------------------------------------------------------------
// Elastodynamics PINN residual on MI455X (gfx1250, wave32).
// Second-order forward-mode jet (value + 4 tangents + 10 Hessian pairs = 15
// channels) propagated through the MLP with V_WMMA_F32_16X16X4_F32.
// One wave = 16 points. 4 waves / workgroup. State lives in LDS planes.
// v2: branchless tanh/sech^2 (no exec-divergent OCML tanh), packed aux planes
//     ((u2,-2vu2) float2, tangents interleaved x4 for 2addr LDS loads).
// ---------------------------------------------------------------------------

#define WAVES   4
#define SROW    52            // state plane row stride (floats); 52*m distinct banks mod 64
#define PLANE   (16 * SROW)   // 832 floats / plane
#define NPLANES 21            // 0:val 1..4:tangents 5..14:H pairs 15..16:(u2,mm)x2 17..20:ZT4
#define WROW    72            // weight LDS row stride; 2*72 % 64 == 16 -> conflict-free B frags
#define WSZ     (52 * WROW)

typedef float v2f __attribute__((ext_vector_type(2)));
typedef float v8f __attribute__((ext_vector_type(8)));

__device__ __forceinline__ v8f wmma4(v2f a, v2f b, v8f c) {
  // D = A(16x4 f32) * B(4x16 f32) + C(16x16 f32)
  return __builtin_amdgcn_wmma_f32_16x16x4_f32(false, a, false, b, (short)0, c,
                                               false, false);
}

// Branchless tanh + sech^2: v = tanh(x), u2 = 1 - v^2, mm = -2*v*u2.
__device__ __forceinline__ void tanh_jet(float x, float& v, float& u2, float& mm) {
  float ax  = __builtin_fabsf(x);
  float e   = __expf(-2.f * ax);                    // v_exp_f32 path, no branches
  float rcp = __builtin_amdgcn_rcpf(1.f + e);
  v  = __builtin_copysignf((1.f - e) * rcp, x);     // v_bfi_b32
  u2 = 4.f * e * rcp * rcp;                         // sech^2, sign-independent
  mm = -2.f * v * u2;
}

// 16x52 (A, LDS plane) @ 52x(16*nTiles) (B, LDS padded weights)
__device__ __forceinline__ void gemm52(const float* __restrict__ Ap,
                                       const float* __restrict__ Wl,
                                       v8f* acc, int nTiles,
                                       int mA, int kOff, int nB) {
  #pragma unroll
  for (int k = 0; k < 52; k += 4) {
    v2f a;
    a.x = Ap[mA * SROW + k + kOff];       // lane<16 -> K=k,k+1 ; lane>=16 -> K=k+2,k+3
    a.y = Ap[mA * SROW + k + kOff + 1];
    #pragma unroll
    for (int nt = 0; nt < 4; ++nt) {
      if (nt < nTiles) {
        int kb = k + kOff;
        v2f b;
        b.x = Wl[kb * WROW + nt * 16 + nB];
        b.y = Wl[(kb + 1) * WROW + nt * 16 + nB];
        acc[nt] = wmma4(a, b, acc[nt]);
      }
    }
  }
}

__device__ __forceinline__ void stage_weights(const float* __restrict__ W,
                                              const float* __restrict__ b,
                                              int din, int dout,
                                              float* ldsW, float* ldsB, int tid) {
  for (int i = tid; i < WSZ; i += WAVES * 32) {
    int r = i / WROW, c = i - r * WROW;
    ldsW[i] = (r < din && c < dout) ? W[r * dout + c] : 0.f;
  }
  for (int i = tid; i < 64; i += WAVES * 32)
    ldsB[i] = (i < dout) ? b[i] : 0.f;
}

__global__ __launch_bounds__(WAVES * 32)
void elasto_jet_kernel(const float* __restrict__ x, const float* __restrict__ y,
                       const float* __restrict__ z, const float* __restrict__ t,
                       const float* __restrict__ W1, const float* __restrict__ b1,
                       const float* __restrict__ W2, const float* __restrict__ b2,
                       const float* __restrict__ W3, const float* __restrict__ b3,
                       const float* __restrict__ W4, const float* __restrict__ b4,
                       const float* __restrict__ W5, const float* __restrict__ b5,
                       const float* __restrict__ W6, const float* __restrict__ b6,
                       const float* __restrict__ W7, const float* __restrict__ b7,
                       const float* __restrict__ c11p, const float* __restrict__ c12p,
                       float* __restrict__ out, int N) {
  extern __shared__ float smem[];
  float* ldsW   = smem;                 // WSZ
  float* ldsB   = smem + WSZ;           // 64
  float* stateA = smem + WSZ + 64;      // WAVES * NPLANES * PLANE

  const int tid    = threadIdx.x;
  const int wave   = tid >> 5;
  const int lane   = tid & 31;
  const int lane16 = lane & 15;
  const int laneHi = lane >> 4;
  const int kOff   = 2 * laneHi;

  float* S    = stateA + wave * (NPLANES * PLANE);
  float* AUX2 = S + 15 * PLANE;   // (u2, mm) interleaved: 2 floats / element
  float* ZT4  = S + 17 * PLANE;   // pre-act tangents interleaved: 4 floats / element

  const int tileBase = (blockIdx.x * WAVES + wave) * 16;

  // ======================= Layer 1: 4 -> 50, tanh =========================
  stage_weights(W1, b1, 4, 50, ldsW, ldsB, tid);
  __syncthreads();
  {
    v8f acc[4];
    #pragma unroll
    for (int nt = 0; nt < 4; ++nt) {
      float bn = ldsB[nt * 16 + lane16];
      #pragma unroll
      for (int q = 0; q < 8; ++q) acc[nt][q] = bn;
    }
    int g = tileBase + lane16;
    v2f a;                      // point coords as A frag: K = {x,y,z,t}
    a.x = laneHi ? z[g] : x[g];
    a.y = laneHi ? t[g] : y[g];
    #pragma unroll
    for (int nt = 0; nt < 4; ++nt) {
      v2f b;
      b.x = ldsW[(kOff) * WROW + nt * 16 + lane16];
      b.y = ldsW[(kOff + 1) * WROW + nt * 16 + lane16];
      acc[nt] = wmma4(a, b, acc[nt]);
    }
    const int PIc[10] = {0, 0, 0, 0, 1, 1, 1, 2, 2, 3};
    const int PJc[10] = {0, 1, 2, 3, 1, 2, 3, 2, 3, 3};
    // Elementwise: input tangents are identity -> z_t_i = W1[i][n], z_H = 0
    #pragma unroll
    for (int nt = 0; nt < 4; ++nt) {
      int n = nt * 16 + lane16;
      bool ok = n < 52;
      float w0 = ldsW[0 * WROW + n];
      float w1 = ldsW[1 * WROW + n];
      float w2 = ldsW[2 * WROW + n];
      float w3 = ldsW[3 * WROW + n];
      #pragma unroll
      for (int q = 0; q < 8; ++q) {
        int m = q + 8 * laneHi;
        float vv, u2, mm;
        tanh_jet(acc[nt][q], vv, u2, mm);
        if (ok) {
          int idx = m * SROW + n;
          float ws[4] = {w0, w1, w2, w3};
          S[0 * PLANE + idx] = vv;
          S[1 * PLANE + idx] = u2 * w0;
          S[2 * PLANE + idx] = u2 * w1;
          S[3 * PLANE + idx] = u2 * w2;
          S[4 * PLANE + idx] = u2 * w3;
          #pragma unroll
          for (int p = 0; p < 10; ++p)
            S[(5 + p) * PLANE + idx] = mm * ws[PIc[p]] * ws[PJc[p]];
        }
      }
    }
  }

  // =================== Layers 2..6: 50 -> 50, tanh ========================
  const float* Wmid[5] = {W2, W3, W4, W5, W6};
  const float* bmid[5] = {b2, b3, b4, b5, b6};
  for (int l = 0; l < 5; ++l) {
    __syncthreads();                       // all waves done reading previous weights
    stage_weights(Wmid[l], bmid[l], 50, 50, ldsW, ldsB, tid);
    __syncthreads();

    // ---- value channel ----
    {
      v8f acc[4];
      #pragma unroll
      for (int nt = 0; nt < 4; ++nt) {
        float bn = ldsB[nt * 16 + lane16];
        #pragma unroll
        for (int q = 0; q < 8; ++q) acc[nt][q] = bn;
      }
      gemm52(S + 0 * PLANE, ldsW, acc, 4, lane16, kOff, lane16);
      #pragma unroll
      for (int nt = 0; nt < 4; ++nt) {
        int n = nt * 16 + lane16;
        bool ok = n < 52;
        #pragma unroll
        for (int q = 0; q < 8; ++q) {
          int m = q + 8 * laneHi;
          float vv, u2, mm;
          tanh_jet(acc[nt][q], vv, u2, mm);
          if (ok) {
            int idx = m * SROW + n;
            S[0 * PLANE + idx] = vv;            // in place: plane fully consumed
            v2f um; um.x = u2; um.y = mm;
            *(v2f*)&AUX2[idx * 2] = um;         // one ds_store_b64
          }
        }
      }
    }
    // ---- tangent channels ----
    #pragma unroll
    for (int i = 0; i < 4; ++i) {
      v8f acc[4];
      #pragma unroll
      for (int nt = 0; nt < 4; ++nt)
        #pragma unroll
        for (int q = 0; q < 8; ++q) acc[nt][q] = 0.f;
      gemm52(S + (1 + i) * PLANE, ldsW, acc, 4, lane16, kOff, lane16);
      #pragma unroll
      for (int nt = 0; nt < 4; ++nt) {
        int n = nt * 16 + lane16;
        bool ok = n < 52;
        #pragma unroll
        for (int q = 0; q < 8; ++q) {
          int m = q + 8 * laneHi;
          float zz = acc[nt][q];
          if (ok) {
            int idx = m * SROW + n;
            ZT4[idx * 4 + i] = zz;              // pre-activation tangent
            float u2 = AUX2[idx * 2];
            S[(1 + i) * PLANE + idx] = u2 * zz; // activated tangent
          }
        }
      }
    }
    // ---- Hessian channels (unrolled: pi/pj constant -> 2addr LDS loads) ----
    const int PIc[10] = {0, 0, 0, 0, 1, 1, 1, 2, 2, 3};
    const int PJc[10] = {0, 1, 2, 3, 1, 2, 3, 2, 3, 3};
    #pragma unroll
    for (int p = 0; p < 10; ++p) {
      v8f acc[4];
      #pragma unroll
      for (int nt = 0; nt < 4; ++nt)
        #pragma unroll
        for (int q = 0; q < 8; ++q) acc[nt][q] = 0.f;
      gemm52(S + (5 + p) * PLANE, ldsW, acc, 4, lane16, kOff, lane16);
      const int pi = PIc[p], pj = PJc[p];
      #pragma unroll
      for (int nt = 0; nt < 4; ++nt) {
        int n = nt * 16 + lane16;
        bool ok = n < 52;
        #pragma unroll
        for (int q = 0; q < 8; ++q) {
          int m = q + 8 * laneHi;
          float zz = acc[nt][q];
          if (ok) {
            int idx = m * SROW + n;
            v2f um = *(const v2f*)&AUX2[idx * 2];            // (u2, mm) b64
            float zi = ZT4[idx * 4 + pi];                    // const offsets ->
            float zj = ZT4[idx * 4 + pj];                    // ds_load_2addr
            S[(5 + p) * PLANE + idx] = um.x * zz + um.y * zi * zj;
          }
        }
      }
    }
  }

  // ===================== Layer 7: 50 -> 3, linear =========================
  __syncthreads();
  stage_weights(W7, b7, 50, 3, ldsW, ldsB, tid);
  __syncthreads();
  {
    // value
    v8f acc[1];
    {
      float bn = ldsB[lane16];
      #pragma unroll
      for (int q = 0; q < 8; ++q) acc[0][q] = bn;
    }
    gemm52(S + 0 * PLANE, ldsW, acc, 1, lane16, kOff, lane16);
    #pragma unroll
    for (int q = 0; q < 8; ++q) {
      int m = q + 8 * laneHi;
      S[0 * PLANE + m * SROW + lane16] = acc[0][q];
    }
    // Hessian channels (tangents/J unused by the reference outputs)
    for (int p = 0; p < 10; ++p) {
      v8f hacc[1];
      #pragma unroll
      for (int q = 0; q < 8; ++q) hacc[0][q] = 0.f;
      gemm52(S + (5 + p) * PLANE, ldsW, hacc, 1, lane16, kOff, lane16);
      #pragma unroll
      for (int q = 0; q < 8; ++q) {
        int m = q + 8 * laneHi;
        S[(5 + p) * PLANE + m * SROW + lane16] = hacc[0][q];
      }
    }
  }

  // ===================== Combine + write outputs ==========================
  if (lane < 16) {
    int m = lane;
    #define LDV(pl, col) S[(pl) * PLANE + m * SROW + (col)]
    float mx = LDV(0, 0), my = LDV(0, 1), mz = LDV(0, 2);
    // pair planes: (0,0)=5 (0,1)=6 (0,2)=7 (0,3)=8 (1,1)=9 (1,2)=10 (1,3)=11 (2,2)=12 (2,3)=13 (3,3)=14
    float H0p00 = LDV(5, 0),  H1p00 = LDV(5, 1),  H2p00 = LDV(5, 2);
    float H0p01 = LDV(6, 0),  H1p01 = LDV(6, 1);
    float H0p02 = LDV(7, 0),  H2p02 = LDV(7, 2);
    float H0p11 = LDV(9, 0),  H1p11 = LDV(9, 1),  H2p11 = LDV(9, 2);
    float H1p12 = LDV(10, 1), H2p12 = LDV(10, 2);
    float H0p22 = LDV(12, 0), H1p22 = LDV(12, 1), H2p22 = LDV(12, 2);
    float ax = LDV(14, 0), ay = LDV(14, 1), az = LDV(14, 2);
    #undef LDV

    float c11 = c11p[0], c12 = c12p[0];
    float mu2 = c11 / (1.f + c12);
    float lam = c11 * c12 / ((1.f + c12) * (1.f - 2.f * c12));

    float lqx  = mu2 * H0p00 + lam * (H0p00 + H1p01);
    float lqxy = mu2 * (H0p01 + H1p00);
    float lqyx = mu2 * (H0p11 + H1p01);
    float lqyy = mu2 * H1p11 + lam * (H0p01 + H1p11);
    float lqz  = mu2 * H2p22 + lam * (H0p02 + H1p12);
    float lqxz = mu2 * (H0p02 + H2p00);
    float lqzx = mu2 * (H0p22 + H2p02);
    float lqyz = mu2 * (H1p12 + H2p11);
    float lqzy = mu2 * (H1p22 + H2p12);

    float fx = 0.113f * ax - lqx - lqyx - lqzx;
    float fy = 0.113f * ay - lqxy - lqyy - lqzy;
    float fz = 0.113f * az - lqz - lqxz - lqyz - 1.1f;

    int g = tileBase + m;
    out[0 * N + g] = mx;
    out[1 * N + g] = my;
    out[2 * N + g] = mz;
    out[3 * N + g] = fx;
    out[4 * N + g] = fy;
    out[5 * N + g] = fz;
  }
}

extern "C" void kernel_launch(void* const* d_in, const int* in_sizes, int n_in,
                              void* d_out, int out_size, void* d_ws, size_t ws_size,
                              hipStream_t stream) {
  const float* x  = (const float*)d_in[0];
  const float* y  = (const float*)d_in[1];
  const float* z  = (const float*)d_in[2];
  const float* t  = (const float*)d_in[3];
  const float* W1 = (const float*)d_in[4];
  const float* b1 = (const float*)d_in[5];
  const float* W2 = (const float*)d_in[6];
  const float* b2 = (const float*)d_in[7];
  const float* W3 = (const float*)d_in[8];
  const float* b3 = (const float*)d_in[9];
  const float* W4 = (const float*)d_in[10];
  const float* b4 = (const float*)d_in[11];
  const float* W5 = (const float*)d_in[12];
  const float* b5 = (const float*)d_in[13];
  const float* W6 = (const float*)d_in[14];
  const float* b6 = (const float*)d_in[15];
  const float* W7 = (const float*)d_in[16];
  const float* b7 = (const float*)d_in[17];
  const float* c11 = (const float*)d_in[18];
  const float* c12 = (const float*)d_in[19];
  float* out = (float*)d_out;

  int N = in_sizes[0];                       // 131072
  int blocks = N / (16 * WAVES);             // 2048
  size_t shmem = (size_t)(WSZ + 64 + WAVES * NPLANES * PLANE) * sizeof(float); // ~295 KB

  elasto_jet_kernel<<<blocks, WAVES * 32, shmem, stream>>>(
      x, y, z, t, W1, b1, W2, b2, W3, b3, W4, b4, W5, b5, W6, b6, W7, b7,
      c11, c12, out, N);
}